// CropAndResize_16630113370849
// MI455X (gfx1250) — compile-verified
//
#include <hip/hip_runtime.h>
#include <hip/hip_bf16.h>

// Crop-and-resize, NCHW, bilinear, extrapolation = 0.
// image: (8, 256, 100, 152) f32; boxes: (1000, 4) f32; box_ind: (1000,) i32
// out:   (1000, 256, 14, 14) f32
//
// MI455X reasoning: 200.7 MB of output writes -> ~8.6 us HBM floor; the 124.5 MB
// image fits in the 192 MB L2 so gathers are L2 hits after the cold pass.
// => write-bound. Use non-temporal stores for the (never re-read) output so it
// does not evict the L2-resident image. Hoist all index/weight math out of the
// 256-channel inner loop. Box coords staged to LDS via the gfx1250 async
// load-to-LDS path (ASYNCcnt).

namespace {
constexpr int CROP  = 14;
constexpr int CH    = 256;
constexpr int IMG_H = 100;
constexpr int IMG_W = 152;
constexpr int HW    = IMG_H * IMG_W;          // 15200
constexpr int PLANE = CROP * CROP;            // 196
constexpr float EXTRAP = 0.0f;
}

__global__ __launch_bounds__(224) void CropAndResize_16630113370849_kernel(
    const float* __restrict__ image,
    const float* __restrict__ boxes,
    const int*   __restrict__ box_ind,
    float* __restrict__ out)
{
    __shared__ float s_box[4];

    const int n   = blockIdx.x;
    const int tid = threadIdx.x;

    // Stage this box's 4 coords into LDS with the CDNA5 async copy engine.
    // One lane issues a 16B async global->LDS load (tracked by ASYNCcnt);
    // the issuing wave waits on ASYNCcnt, then the workgroup barrier makes
    // the LDS data visible to all 7 waves.
    if (tid == 0) {
        unsigned lds_addr = (unsigned)(uintptr_t)(s_box); // low 32b = LDS offset
        unsigned voff     = 0u;
        const float* bp   = boxes + 4 * n;                // uniform -> SGPR pair
        asm volatile("global_load_async_to_lds_b128 %0, %1, %2"
                     :
                     : "v"(lds_addr), "v"(voff), "s"(bp)
                     : "memory");
        asm volatile("s_wait_asynccnt 0" ::: "memory");
    }
    __syncthreads();

    if (tid >= PLANE) return;

    const int y = tid / CROP;
    const int x = tid - y * CROP;

    const float y1 = s_box[0];
    const float x1 = s_box[1];
    const float y2 = s_box[2];
    const float x2 = s_box[3];

    // Same arithmetic as the reference (matches its rounding behavior).
    const float h_scale = (y2 - y1) * (float)(IMG_H - 1) / (float)(CROP - 1);
    const float w_scale = (x2 - x1) * (float)(IMG_W - 1) / (float)(CROP - 1);
    const float in_y = y1 * (float)(IMG_H - 1) + (float)y * h_scale;
    const float in_x = x1 * (float)(IMG_W - 1) + (float)x * w_scale;

    const bool valid = (in_y >= 0.0f) && (in_y <= (float)(IMG_H - 1)) &&
                       (in_x >= 0.0f) && (in_x <= (float)(IMG_W - 1));

    const float top  = floorf(in_y);
    const float left = floorf(in_x);
    const float y_lerp = in_y - top;
    const float x_lerp = in_x - left;

    const int ti = (int)fminf(fmaxf(top,         0.0f), (float)(IMG_H - 1));
    const int bi = (int)fminf(fmaxf(top  + 1.0f, 0.0f), (float)(IMG_H - 1));
    const int li = (int)fminf(fmaxf(left,        0.0f), (float)(IMG_W - 1));
    const int ri = (int)fminf(fmaxf(left + 1.0f, 0.0f), (float)(IMG_W - 1));

    const int b = box_ind[n];
    const float* __restrict__ base = image + (size_t)b * (size_t)(CH * HW);

    const int oTL = ti * IMG_W + li;
    const int oTR = ti * IMG_W + ri;
    const int oBL = bi * IMG_W + li;
    const int oBR = bi * IMG_W + ri;

    float* __restrict__ obase = out + (size_t)n * (size_t)(CH * PLANE) + tid;

    // 256-channel loop: 4 L2-resident gathers + 3 FMAs + 1 NT store per iter.
    #pragma unroll 4
    for (int c = 0; c < CH; ++c) {
        const float* __restrict__ pc = base + (size_t)c * HW;
        const float tl = pc[oTL];
        const float tr = pc[oTR];
        const float bl = pc[oBL];
        const float br = pc[oBR];
        const float top_v = tl + (tr - tl) * x_lerp;
        const float bot_v = bl + (br - bl) * x_lerp;
        float v = top_v + (bot_v - top_v) * y_lerp;
        v = valid ? v : EXTRAP;
        // Non-temporal: keep the image resident in the 192 MB L2.
        __builtin_nontemporal_store(v, obase + (size_t)c * PLANE);
    }
}

extern "C" void kernel_launch(void* const* d_in, const int* in_sizes, int n_in,
                              void* d_out, int out_size, void* d_ws, size_t ws_size,
                              hipStream_t stream) {
    (void)n_in; (void)out_size; (void)d_ws; (void)ws_size;
    const float* image   = (const float*)d_in[0];
    const float* boxes   = (const float*)d_in[1];
    const int*   box_ind = (const int*)d_in[2];
    float*       out     = (float*)d_out;

    const int nbox = in_sizes[1] / 4;   // 1000
    CropAndResize_16630113370849_kernel<<<nbox, 224, 0, stream>>>(
        image, boxes, box_ind, out);
}